// MultiLayerGRUParallel_43379169689746
// MI455X (gfx1250) — compile-verified
//
#include <hip/hip_runtime.h>
#include <math.h>

// Problem constants (from reference)
#define BB   32
#define SS   4096
#define HH   128
#define OUTN 128
#define NC   64            // number of scan chunks
#define CHUNK (SS/NC)      // 64 timesteps per chunk
#define KSL  2048          // split-K slice length for output GEMM
#define NSLICES ((SS*HH)/KSL)  // 256 slices

typedef float v2f __attribute__((ext_vector_type(2)));
typedef float v8f __attribute__((ext_vector_type(8)));
typedef unsigned int u32x4 __attribute__((ext_vector_type(4)));
typedef int i32x4 __attribute__((ext_vector_type(4)));
typedef int i32x8 __attribute__((ext_vector_type(8)));

#define NEG_BIG (-3.0e38f)

__device__ __forceinline__ float softplusf(float x) {
  // log(1+e^x) = max(x,0) + log1p(exp(-|x|))  (stable)
  return fmaxf(x, 0.f) + log1pf(__expf(-fabsf(x)));
}
__device__ __forceinline__ float logaddexpf_(float a, float b) {
  float mx = fmaxf(a, b), mn = fminf(a, b);
  if (mx <= NEG_BIG) return mx;            // avoid (-inf) - (-inf)
  return mx + log1pf(__expf(mn - mx));
}
__device__ __forceinline__ float log_g(float x) {
  // x>=0: log(relu(x)+0.5)=log(x+0.5) ; x<0: -softplus(-x)
  return (x >= 0.f) ? __logf(x + 0.5f) : -softplusf(-x);
}

// ---------------- TDM: 2-D tile Global -> LDS (D# per CDNA5 ISA)
// data_size = 4 bytes (code 2). Groups 2/3 zero (2-D tensor; higher tile dims
// unused). This toolchain exposes the 6-arg builtin form:
//   (u32x4 g0, i32x8 g1, i32x4, i32x4, i32x8, i32 cpol)
__device__ __forceinline__ void tdm_load_2d_f32(const void* gaddr, unsigned lds_off,
                                                unsigned row_elems, unsigned rows,
                                                unsigned row_stride_elems) {
  unsigned long long ga = (unsigned long long)gaddr;
  u32x4 g0;
  g0[0] = 1u;                                   // count=1 (valid user descriptor)
  g0[1] = lds_off;                              // lds_addr (bytes)
  g0[2] = (unsigned)(ga & 0xFFFFFFFFu);         // global_addr[31:0]
  g0[3] = (unsigned)((ga >> 32) & 0x1FFFFFFu)   // global_addr[56:32]
          | (2u << 30);                         // type = 2 ("image")
  i32x8 g1;
  g1[0] = (int)(2u << 16);                      // workgroup_mask=0, data_size=2 (4B)
  g1[1] = (int)((row_elems & 0xFFFFu) << 16);   // tensor_dim0[15:0] in dw1[31:16]
  g1[2] = (int)(((row_elems >> 16) & 0xFFFFu) | ((rows & 0xFFFFu) << 16));   // dim0 hi | dim1 lo
  g1[3] = (int)(((rows >> 16) & 0xFFFFu) | ((row_elems & 0xFFFFu) << 16));   // dim1 hi | tile_dim0
  g1[4] = (int)(rows & 0xFFFFu);                // tile_dim1 | tile_dim2=0
  g1[5] = (int)row_stride_elems;                // tensor_dim0_stride[31:0]
  g1[6] = 0;                                    // stride0 hi | stride1 lo
  g1[7] = 0;                                    // stride1 hi
  i32x4 gz = {0, 0, 0, 0};
  i32x8 gz8 = {0, 0, 0, 0, 0, 0, 0, 0};
  __builtin_amdgcn_tensor_load_to_lds(g0, g1, gz, gz, gz8, 0);
}

// ---------------- Layer 0 pre-activations (din = 1, pure broadcast) --------
__global__ void layer0_pre(const float* __restrict__ x,
                           const float* __restrict__ Wz, const float* __restrict__ bz,
                           const float* __restrict__ Wh, const float* __restrict__ bh,
                           float* __restrict__ Kb, float* __restrict__ THb) {
  int h = threadIdx.x;              // 0..127
  int bt = blockIdx.x;              // 0..B*S-1
  float xv = x[bt];
  size_t idx = (size_t)bt * HH + h;
  Kb[idx]  = xv * Wz[h] + bz[h];
  THb[idx] = xv * Wh[h] + bh[h];
}

// ---------------- Inner-layer GEMMs: Y = Hin @ W.T + bias (f32 WMMA) -------
// One M-tile (16 rows) per block; 8 waves cover the 8 N-tiles. The whole
// 128x128 weight (64KB) + the 16x128 A tile (8KB) are staged into LDS with a
// single TDM tensor_load_to_lds each, then all fragments come from LDS.
// blockIdx.y: 0 -> (Wz,bz)->Kb ; 1 -> (Wh,bh)->THb
__global__ void gemm128_wmma(const float* __restrict__ Hin,
                             const float* __restrict__ Wz, const float* __restrict__ bz,
                             const float* __restrict__ Wh, const float* __restrict__ bh,
                             float* __restrict__ Kb, float* __restrict__ THb) {
  __shared__ float Wlds[HH * HH];   // 64 KB
  __shared__ float Alds[16 * HH];   // 8 KB

  const float* W    = blockIdx.y ? Wh  : Wz;
  const float* bias = blockIdx.y ? bh  : bz;
  float*       Y    = blockIdx.y ? THb : Kb;

  int wave = threadIdx.x >> 5;      // 0..7 == N tile
  int lane = threadIdx.x & 31;
  int mt   = blockIdx.x;            // 0..8191 M tiles
  int m    = lane & 15;
  int hi   = lane >> 4;

  if (wave == 0) {                  // TDM issues once per wave (EXEC ignored)
    tdm_load_2d_f32(W, (unsigned)(size_t)&Wlds[0], HH, HH, HH);
    tdm_load_2d_f32(Hin + (size_t)mt * 16 * HH, (unsigned)(size_t)&Alds[0], HH, 16, HH);
    __builtin_amdgcn_s_wait_tensorcnt(0);
  }
  __syncthreads();

  // A (16x4 f32): lanes 0-15 rows M, VGPR0/1 = K 2*hi,2*hi+1
  const float* arow = &Alds[m * HH + 2 * hi];
  // B (4x16 f32): B[k][n] = W[n][k]; lane carries n, hi selects K pair
  const float* brow = &Wlds[(wave * 16 + m) * HH + 2 * hi];

  v8f acc = {};
#pragma unroll 8
  for (int kk = 0; kk < HH; kk += 4) {
    v2f a  = *(const v2f*)(arow + kk);
    v2f bf = *(const v2f*)(brow + kk);
    acc = __builtin_amdgcn_wmma_f32_16x16x4_f32(
        /*neg_a=*/false, a, /*neg_b=*/false, bf,
        /*c_mod=*/(short)0, acc, /*reuse_a=*/false, /*reuse_b=*/false);
  }
  float bv = bias[wave * 16 + m];
#pragma unroll
  for (int j = 0; j < 8; ++j) {
    size_t row = (size_t)(mt * 16 + j + 8 * hi);   // C/D: VGPR j -> M=j (+8 for hi lanes)
    Y[row * HH + wave * 16 + m] = acc[j] + bv;
  }
}

// ---------------- Scan pass 1: per-chunk local logaddexp scan --------------
// block = 128 threads (one per h), grid = (B, NC). In-place: KA := a_star,
// THR := local inclusive scan value. Per-timestep inclusive prefix over H via LDS.
__global__ void scan_pass1(float* __restrict__ KA, float* __restrict__ THR,
                           float* __restrict__ Ctot) {
  __shared__ float s[HH];
  int h = threadIdx.x;
  int b = blockIdx.x;
  int c = blockIdx.y;
  int t0 = c * CHUNK;
  float lr = NEG_BIG;
  for (int i = 0; i < CHUNK; ++i) {
    int t = t0 + i;
    size_t idx = ((size_t)(b * SS + t)) * HH + h;
    float k   = KA[idx];
    float thp = THR[idx];
    float lz  = -softplusf(-k);
    float lc  = -softplusf(k);
    float lth = log_g(thp);
    // inclusive prefix-sum of lc over h (reference cumsums over axis=-1 == H)
    s[h] = lc;
    __syncthreads();
    for (int off = 1; off < HH; off <<= 1) {
      float v = (h >= off) ? s[h - off] : 0.f;
      __syncthreads();
      s[h] += v;
      __syncthreads();
    }
    float astar = s[h];
    __syncthreads();                       // protect s[] before next iteration
    float u = lz + lth - astar;            // log_values[t+1] - a_star[t+1]
    lr = logaddexpf_(lr, u);
    KA[idx]  = astar;                      // store a_star for pass 3
    THR[idx] = lr;                         // store local inclusive scan
  }
  Ctot[((size_t)(b * HH + h)) * NC + c] = lr;   // chunk total
}

// ---------------- Scan pass 2: exclusive logaddexp scan over chunk totals ---
__global__ void scan_pass2(const float* __restrict__ Ctot, float* __restrict__ Cpre) {
  int id = blockIdx.x * blockDim.x + threadIdx.x;   // (b,h) channel
  if (id >= BB * HH) return;
  size_t base = (size_t)id * NC;
  float P = -0.69314718055994531f;                  // log(0.5): the t=0 term
  for (int c = 0; c < NC; ++c) {
    Cpre[base + c] = P;
    P = logaddexpf_(P, Ctot[base + c]);
  }
}

// ---------------- Scan pass 3: finalize h = sigmoid(exp(log_h)) ------------
__global__ void scan_pass3(const float* __restrict__ Abuf, const float* __restrict__ Rbuf,
                           const float* __restrict__ Cpre, float* __restrict__ hout) {
  int h  = threadIdx.x;
  int bt = blockIdx.x;
  int b  = bt >> 12;            // /S
  int t  = bt & (SS - 1);
  int c  = t / CHUNK;
  size_t idx = (size_t)bt * HH + h;
  float P  = Cpre[((size_t)(b * HH + h)) * NC + c];
  float lh = Abuf[idx] + logaddexpf_(P, Rbuf[idx]);
  float e  = __expf(lh);
  hout[idx] = 1.f / (1.f + __expf(-e));
}

// ---------------- Output projection: split-K f32 WMMA ----------------------
// out(32x128) = Hfin(32 x 524288) @ Wout.T ; 16 waves/block = all 16 tiles,
// grid.x = 256 K-slices; Wout streamed once (268MB, pure HBM-bound) with
// prefetch; deterministic fixed-order reduction afterwards.
__global__ void outgemm_wmma(const float* __restrict__ Hfin,
                             const float* __restrict__ Wout,
                             float* __restrict__ partial) {
  int wave = threadIdx.x >> 5;          // 0..15
  int lane = threadIdx.x & 31;
  int slice = blockIdx.x;               // 0..NSLICES-1
  int nt = wave & 7;                    // 8 N tiles
  int mt = wave >> 3;                   // 2 M tiles
  int m  = lane & 15;
  int hi = lane >> 4;
  size_t k0 = (size_t)slice * KSL;
  size_t ldk = (size_t)SS * HH;         // 524288
  const float* arow = Hfin + ((size_t)(mt * 16 + m)) * ldk + k0 + 2 * hi;
  const float* brow = Wout + ((size_t)(nt * 16 + m)) * ldk + k0 + 2 * hi;
  v8f acc = {};
  for (int kk = 0; kk < KSL; kk += 4) {
    __builtin_prefetch(arow + kk + 128, 0, 1);
    __builtin_prefetch(brow + kk + 128, 0, 1);
    v2f a  = *(const v2f*)(arow + kk);
    v2f bf = *(const v2f*)(brow + kk);
    acc = __builtin_amdgcn_wmma_f32_16x16x4_f32(false, a, false, bf,
                                                (short)0, acc, false, false);
  }
  float* p = partial + (size_t)slice * (BB * OUTN);
#pragma unroll
  for (int j = 0; j < 8; ++j) {
    int row = mt * 16 + j + 8 * hi;
    p[row * OUTN + nt * 16 + m] = acc[j];
  }
}

__global__ void out_reduce(const float* __restrict__ partial,
                           const float* __restrict__ bout, float* __restrict__ out) {
  int i = blockIdx.x * blockDim.x + threadIdx.x;
  if (i >= BB * OUTN) return;
  float acc = bout[i & (OUTN - 1)];
  for (int s = 0; s < NSLICES; ++s)         // fixed order -> deterministic
    acc += partial[(size_t)s * (BB * OUTN) + i];
  out[i] = acc;
}

// ---------------- Orchestration --------------------------------------------
extern "C" void kernel_launch(void* const* d_in, const int* in_sizes, int n_in,
                              void* d_out, int out_size, void* d_ws, size_t ws_size,
                              hipStream_t stream) {
  (void)in_sizes; (void)n_in; (void)out_size; (void)ws_size;
  const float* x  = (const float*)d_in[0];
  const float* Wz[3] = {(const float*)d_in[1], (const float*)d_in[5],  (const float*)d_in[9]};
  const float* bz[3] = {(const float*)d_in[2], (const float*)d_in[6],  (const float*)d_in[10]};
  const float* Wh[3] = {(const float*)d_in[3], (const float*)d_in[7],  (const float*)d_in[11]};
  const float* bh[3] = {(const float*)d_in[4], (const float*)d_in[8],  (const float*)d_in[12]};
  const float* Wout  = (const float*)d_in[13];
  const float* bout  = (const float*)d_in[14];
  float* out = (float*)d_out;

  // Workspace carve-up (~207 MB): h | K/a_star | TH/local-scan | chunk totals
  // | chunk prefixes | split-K partials
  size_t BSH = (size_t)BB * SS * HH;
  float* hbuf    = (float*)d_ws;
  float* Kb      = hbuf + BSH;
  float* THb     = Kb + BSH;
  float* Ctot    = THb + BSH;
  float* Cpre    = Ctot + (size_t)BB * HH * NC;
  float* partial = Cpre + (size_t)BB * HH * NC;

  dim3 scan1Grid(BB, NC);
  for (int l = 0; l < 3; ++l) {
    if (l == 0) {
      layer0_pre<<<BB * SS, HH, 0, stream>>>(x, Wz[0], bz[0], Wh[0], bh[0], Kb, THb);
    } else {
      dim3 g(BB * SS / 16, 2);   // 8192 M-tiles x {Wz,Wh}
      gemm128_wmma<<<g, 256, 0, stream>>>(hbuf, Wz[l], bz[l], Wh[l], bh[l], Kb, THb);
    }
    scan_pass1<<<scan1Grid, HH, 0, stream>>>(Kb, THb, Ctot);
    scan_pass2<<<(BB * HH + 255) / 256, 256, 0, stream>>>(Ctot, Cpre);
    scan_pass3<<<BB * SS, HH, 0, stream>>>(Kb, THb, Cpre, hbuf);
  }
  outgemm_wmma<<<NSLICES, 512, 0, stream>>>(hbuf, Wout, partial);
  out_reduce<<<(BB * OUTN + 255) / 256, 256, 0, stream>>>(partial, bout, out);
}